// Loss_88467736363518
// MI455X (gfx1250) — compile-verified
//
#include <hip/hip_runtime.h>

// ---------------------------------------------------------------------------
// YOLO-v1 style loss on MI455X (gfx1250, wave32).
// Bandwidth-bound: ~112 MB read -> ~4.8 us floor at 23.3 TB/s, so the design
// goals are (a) perfectly coalesced global traffic via LDS staging,
// (b) minimal VALU per cell (compute budget is comparable to the BW floor),
// (c) deterministic atomic-free reduction using the CDNA5 WMMA unit.
//
// LDS layout: each cell's 30 floats padded to 34 -> with 64 LDS banks,
// stride-34 float2 accesses are bank-conflict-free, so both staging writes
// and compute reads are b64 DS ops.
// Wave reduction: one v_wmma_f32_16x16x4_f32 (B = ones) folds 32 lane
// partials to half-wave sums, then a ds_swizzle SWAPX16 finishes the wave.
// ---------------------------------------------------------------------------

typedef float v2f __attribute__((ext_vector_type(2)));
typedef float v8f __attribute__((ext_vector_type(8)));

#define GRID_NUM 28
#define CELLS_PER_IMG (GRID_NUM * GRID_NUM)
#define GS 8.0f         // 224 / 28
#define HALF_IMG 112.0f
#define LAMBDA_COORD 5.0f
#define LAMBDA_NOOBJ 0.5f
#define NUM_CLASSES 20
#define EPS_IOU 1e-12f

#define BLOCK 256
#define PRED_PAD 34     // 30 floats/cell padded to 34: stride-34 float2 is
#define PRED_PAD2 17    // conflict-free across 64 LDS banks

// lane <-> lane^16 within wave32 via DS_SWIZZLE_B32 group-of-32 mode:
// offset = xor_mask(0x10)<<10 | or(0)<<5 | and(0x1f)  = 0x401F  (SWAPX16)
__device__ __forceinline__ float swap16(float x) {
    return __int_as_float(__builtin_amdgcn_ds_swizzle(__float_as_int(x), 0x401F));
}

// Wave32 sum reduction using one V_WMMA_F32_16X16X4_F32.
// A: lane value in K0/K2 slot (a.x), zeros in K1/K3 (a.y). B: all ones.
// D[m][n] = s[m] + s[m+16]; summing the 8 D regs yields the half-wave total
// in every lane; swap16 adds the other half. All lanes return the wave total.
// EXEC must be all-1s at the call site (no early returns before this).
__device__ __forceinline__ float wave_reduce_wmma(float s) {
    v2f a; a.x = s;    a.y = 0.0f;
    v2f b; b.x = 1.0f; b.y = 1.0f;
    v8f c = {};
    c = __builtin_amdgcn_wmma_f32_16x16x4_f32(
            /*neg_a=*/false, a, /*neg_b=*/false, b,
            /*c_mod=*/(short)0, c, /*reuse_a=*/false, /*reuse_b=*/false);
    float e01 = c[0] + c[1], e23 = c[2] + c[3];
    float e45 = c[4] + c[5], e67 = c[6] + c[7];
    float e = (e01 + e23) + (e45 + e67);
    return e + swap16(e);
}

__global__ __launch_bounds__(BLOCK) void yolo_loss_main(
    const float* __restrict__ y_pred,   // [nCells, 30]
    const float* __restrict__ y_true,   // [nCells, 5]
    float* __restrict__ partials,       // [gridDim.x]
    int nCells)
{
    __shared__ __align__(16) float lp[BLOCK * PRED_PAD];   // 34816 B
    __shared__ __align__(16) float lt[BLOCK * 5];          //  5120 B
    __shared__ float wpart[BLOCK / 32];

    const int tid       = threadIdx.x;
    const int blockBase = blockIdx.x * BLOCK;           // first cell of block
    const int cells     = min(BLOCK, nCells - blockBase);

    // ---- Stage y_pred: 3840 float2 per block, fully coalesced global b64
    //      loads -> conflict-free ds_store_b64 (padded layout) ----
    const float2* gp2 = reinterpret_cast<const float2*>(
        y_pred + (size_t)blockBase * 30);               // 8B-aligned (30 even)
    float2* lp2 = reinterpret_cast<float2*>(lp);
    const float* gt = y_true + (size_t)blockBase * 5;

    if (cells == BLOCK) {                               // uniform fast path
        #pragma unroll
        for (int k = 0; k < 15; ++k) {
            int m = tid + k * BLOCK;
            float2 v = gp2[m];
            int cell = m / 15;                          // 30 even -> float2
            int j    = m - cell * 15;                   // never splits a cell
            lp2[cell * PRED_PAD2 + j] = v;
        }
        #pragma unroll
        for (int k = 0; k < 5; ++k) {
            int m = tid + k * BLOCK;
            lt[m] = gt[m];
        }
    } else {                                            // guarded tail blocks
        const int elems2 = cells * 15;
        const int elemsT = cells * 5;
        #pragma unroll
        for (int k = 0; k < 15; ++k) {
            int m = tid + k * BLOCK;
            float2 v = make_float2(0.0f, 0.0f);
            if (m < elems2) v = gp2[m];
            int cell = m / 15;
            int j    = m - cell * 15;
            lp2[cell * PRED_PAD2 + j] = v;
        }
        #pragma unroll
        for (int k = 0; k < 5; ++k) {
            int m = tid + k * BLOCK;
            lt[m] = (m < elemsT) ? gt[m] : 0.0f;
        }
    }

    __syncthreads();

    // ---- Per-cell loss ----
    const int   cellIdx = blockBase + tid;
    const bool  valid   = cellIdx < nCells;
    const float2* P2 = reinterpret_cast<const float2*>(lp + tid * PRED_PAD);
    const float*  T  = &lt[tid * 5];

    int rem = cellIdx % CELLS_PER_IMG;
    float fgi = (float)(rem / GRID_NUM);
    float fgj = (float)(rem % GRID_NUM);

    float cls = T[0];
    float tx = T[1], ty = T[2], tw = T[3], th = T[4];

    // first 10 prediction floats: two boxes + confidences
    float2 p01 = P2[0], p23 = P2[1], p45 = P2[2], p67 = P2[3], p89 = P2[4];
    float conf0 = p45.x, conf1 = p89.y;

    // true-box corners
    float tcx = (fgj + tx) * GS, tcy = (fgi + ty) * GS;
    float tw2 = tw * HALF_IMG,  th2 = th * HALF_IMG;
    float bx1 = tcx - tw2, by1 = tcy - th2;
    float bx2 = tcx + tw2, by2 = tcy + th2;
    float area_b = fmaxf(bx2 - bx1, 0.0f) * fmaxf(by2 - by1, 0.0f);

    float bxv[2][4] = { { p01.x, p01.y, p23.x, p23.y },
                        { p45.y, p67.x, p67.y, p89.x } };
    float iou[2];
    #pragma unroll
    for (int k = 0; k < 2; ++k) {
        float px = bxv[k][0], py = bxv[k][1], pw = bxv[k][2], ph = bxv[k][3];
        float cx = (fgj + px) * GS, cy = (fgi + py) * GS;
        float w2 = pw * HALF_IMG,   h2 = ph * HALF_IMG;
        float ax1 = cx - w2, ay1 = cy - h2;
        float ax2 = cx + w2, ay2 = cy + h2;
        float iw = fmaxf(fminf(ax2, bx2) - fmaxf(ax1, bx1), 0.0f);
        float ih = fmaxf(fminf(ay2, by2) - fmaxf(ay1, by1), 0.0f);
        float inter  = iw * ih;
        float area_a = fmaxf(ax2 - ax1, 0.0f) * fmaxf(ay2 - ay1, 0.0f);
        iou[k] = inter / (area_a + area_b - inter + EPS_IOU);
    }

    bool  choose1   = !(iou[0] > iou[1]);
    float conf_pred = choose1 ? conf1 : conf0;
    float conf_true = choose1 ? iou[1] : iou[0];
    float xp        = choose1 ? p45.y : p01.x;
    float yp        = choose1 ? p67.x : p01.y;

    float dc = conf_pred - conf_true;
    float loss_conf_obj = dc * dc;

    float dx = xp - tx, dy = yp - ty;
    float loss_coord = LAMBDA_COORD * (dx * dx + dy * dy);

    // class loss: sum(p^2) - 2*p[cls] + 1  (== sum((p - onehot)^2))
    float s0 = 0.0f, s1 = 0.0f;
    #pragma unroll
    for (int c = 0; c < NUM_CLASSES / 2; ++c) {
        float2 q = P2[5 + c];
        s0 = fmaf(q.x, q.x, s0);
        s1 = fmaf(q.y, q.y, s1);
    }
    int ci  = (int)cls - 1;                 // -1 => one-hot all zeros (unused)
    int cid = (ci < 0) ? 0 : ci;
    float pcls = lp[tid * PRED_PAD + 10 + cid];   // dynamic conflict-free read
    float loss_cls = (s0 + s1) - 2.0f * pcls + 1.0f;

    float loss_noobj = LAMBDA_NOOBJ * (conf0 * conf0 + conf1 * conf1);

    bool  obj  = (cls != 0.0f);
    float cell = obj ? (loss_conf_obj + loss_coord + loss_cls) : loss_noobj;
    if (!valid) cell = 0.0f;

    // ---- Block reduction: WMMA wave reduce, then 8 partials via LDS ----
    float wsum = wave_reduce_wmma(cell);
    if ((tid & 31) == 0) wpart[tid >> 5] = wsum;
    __syncthreads();
    if (tid == 0) {
        float t = 0.0f;
        #pragma unroll
        for (int i = 0; i < BLOCK / 32; ++i) t += wpart[i];
        partials[blockIdx.x] = t;
    }
}

__global__ __launch_bounds__(BLOCK) void yolo_loss_final(
    const float* __restrict__ partials, int n,
    float* __restrict__ out, float invB)
{
    __shared__ float wpart[BLOCK / 32];
    float s = 0.0f;
    for (int i = threadIdx.x; i < n; i += BLOCK) s += partials[i];

    float wsum = wave_reduce_wmma(s);
    if ((threadIdx.x & 31) == 0) wpart[threadIdx.x >> 5] = wsum;
    __syncthreads();
    if (threadIdx.x == 0) {
        float t = 0.0f;
        #pragma unroll
        for (int i = 0; i < BLOCK / 32; ++i) t += wpart[i];
        out[0] = t * invB;
    }
}

extern "C" void kernel_launch(void* const* d_in, const int* in_sizes, int n_in,
                              void* d_out, int out_size, void* d_ws, size_t ws_size,
                              hipStream_t stream) {
    (void)n_in; (void)out_size; (void)ws_size;
    const float* y_pred = (const float*)d_in[0];
    const float* y_true = (const float*)d_in[1];

    const int nPred  = in_sizes[0];          // B * 28 * 28 * 30
    const int nCells = nPred / 30;
    const int B      = nCells / CELLS_PER_IMG;
    const float invB = 1.0f / (float)B;

    const int blocks = (nCells + BLOCK - 1) / BLOCK;   // 3136 for B=1024
    float* partials = (float*)d_ws;

    yolo_loss_main<<<blocks, BLOCK, 0, stream>>>(y_pred, y_true, partials, nCells);
    yolo_loss_final<<<1, BLOCK, 0, stream>>>(partials, blocks, (float*)d_out, invB);
}